// GraphSageLayer_47725676593247
// MI455X (gfx1250) — compile-verified
//
#include <hip/hip_runtime.h>

#define N_NODES 50000
#define N_EDGES 800000
#define D       128
#define M_TILES (N_NODES / 16)   // 3125, exact
#define BN_EPS  1e-5f
#define SLOPE   0.01f

typedef float v2f __attribute__((ext_vector_type(2)));
typedef float v8f __attribute__((ext_vector_type(8)));

// ---------------------------------------------------------------------------
// Kernel 1: zero the atomic accumulators (agg, deg, inv_deg, sum, sumsq)
// ---------------------------------------------------------------------------
__global__ void k_zero(float4* __restrict__ p, long nvec) {
    long i = (long)blockIdx.x * blockDim.x + threadIdx.x;
    if (i < nvec) p[i] = make_float4(0.f, 0.f, 0.f, 0.f);
}

// ---------------------------------------------------------------------------
// Kernel 2: edge scatter-add.  One wave32 per edge; lane l moves 4 floats.
// agg[dst] += h[src]; deg[dst] += 1.  agg fits in L2 -> fast L2 atomics.
// ---------------------------------------------------------------------------
__global__ void k_scatter(const float* __restrict__ h,
                          const int* __restrict__ edge_index,
                          float* __restrict__ agg,
                          float* __restrict__ deg) {
    long gid  = (long)blockIdx.x * blockDim.x + threadIdx.x;
    int  edge = (int)(gid >> 5);
    int  lane = (int)(gid & 31);
    if (edge >= N_EDGES) return;
    int src = edge_index[edge];            // row 0 of [2, E]
    int dst = edge_index[N_EDGES + edge];  // row 1 of [2, E]
    const float4 v = *(const float4*)(h + (size_t)src * D + lane * 4);
    float* a = agg + (size_t)dst * D + lane * 4;
    __hip_atomic_fetch_add(a + 0, v.x, __ATOMIC_RELAXED, __HIP_MEMORY_SCOPE_AGENT);
    __hip_atomic_fetch_add(a + 1, v.y, __ATOMIC_RELAXED, __HIP_MEMORY_SCOPE_AGENT);
    __hip_atomic_fetch_add(a + 2, v.z, __ATOMIC_RELAXED, __HIP_MEMORY_SCOPE_AGENT);
    __hip_atomic_fetch_add(a + 3, v.w, __ATOMIC_RELAXED, __HIP_MEMORY_SCOPE_AGENT);
    if (lane == 0)
        __hip_atomic_fetch_add(deg + dst, 1.0f, __ATOMIC_RELAXED, __HIP_MEMORY_SCOPE_AGENT);
}

// ---------------------------------------------------------------------------
// Kernel 3: inv_deg = 1 / max(deg, 1)
// ---------------------------------------------------------------------------
__global__ void k_invdeg(const float* __restrict__ deg, float* __restrict__ invdeg) {
    int i = blockIdx.x * blockDim.x + threadIdx.x;
    if (i < N_NODES) invdeg[i] = 1.0f / fmaxf(deg[i], 1.0f);
}

// ---------------------------------------------------------------------------
// Kernel 4: fused dual GEMM (fp32 WMMA 16x16x4) + bias + leaky ReLU
//           + BN partial sums.  8 waves/block; wave w owns output columns
//           [16w, 16w+16); block loops over M tiles.
// WMMA fp32 layouts (wave32):
//   A 16x4 : lane&15 = M row; a.x = K(=k+2*(lane>>4)), a.y = K+1
//   B 4x16 : lane&15 = N col; b.x = row K(=k+2*(lane>>4)), b.y = row K+1
//   C 16x16: lane&15 = N col; c[i] = row m0 + i + 8*(lane>>4)
// ---------------------------------------------------------------------------
__global__ void k_gemm_bn(const float* __restrict__ h,
                          const float* __restrict__ agg,
                          const float* __restrict__ invdeg,
                          const float* __restrict__ Wself,
                          const float* __restrict__ Wneigh,
                          const float* __restrict__ bias,
                          float* __restrict__ rst,
                          float* __restrict__ gsum,
                          float* __restrict__ gsumsq) {
    __shared__ float ls[D];
    __shared__ float lsq[D];
    int tid = threadIdx.x;
    if (tid < D) { ls[tid] = 0.f; lsq[tid] = 0.f; }
    __syncthreads();

    const int wave = tid >> 5;
    const int lane = tid & 31;
    const int half = lane >> 4;       // 0 or 1
    const int l15  = lane & 15;
    const int ncol = wave * 16 + l15; // this lane's output column
    const float bv = bias[ncol];

    float acc_s = 0.f, acc_q = 0.f;

    for (int mt = blockIdx.x; mt < M_TILES; mt += gridDim.x) {
        const int m0  = mt * 16;
        const int row = m0 + l15;
        const float rdeg = invdeg[row];
        const float* __restrict__ hrow = h   + (size_t)row * D;
        const float* __restrict__ arow = agg + (size_t)row * D;

        v8f c = {0.f, 0.f, 0.f, 0.f, 0.f, 0.f, 0.f, 0.f};

        #pragma unroll 4
        for (int k = 0; k < D; k += 4) {
            const int ka = k + 2 * half;
            // A fragments: h row and (agg * inv_deg) row
            v2f a_h = *(const v2f*)(hrow + ka);
            v2f a_n;
            a_n.x = arow[ka]     * rdeg;
            a_n.y = arow[ka + 1] * rdeg;
            // B fragments: rows ka, ka+1 of W matrices at column ncol
            v2f b_s, b_n;
            b_s.x = Wself [(size_t)(ka)     * D + ncol];
            b_s.y = Wself [(size_t)(ka + 1) * D + ncol];
            b_n.x = Wneigh[(size_t)(ka)     * D + ncol];
            b_n.y = Wneigh[(size_t)(ka + 1) * D + ncol];
            c = __builtin_amdgcn_wmma_f32_16x16x4_f32(
                    false, a_h, false, b_s, (short)0, c, false, false);
            c = __builtin_amdgcn_wmma_f32_16x16x4_f32(
                    false, a_n, false, b_n, (short)0, c, false, false);
        }

        #pragma unroll
        for (int i = 0; i < 8; ++i) {
            float x = c[i] + bv;
            x = (x > 0.f) ? x : x * SLOPE;       // leaky ReLU
            const int m = m0 + i + 8 * half;
            rst[(size_t)m * D + ncol] = x;
            acc_s += x;
            acc_q += x * x;
        }
    }

    // per-block column reduction in LDS (ds_add_f32), then one global atomic
    __hip_atomic_fetch_add(&ls[ncol],  acc_s, __ATOMIC_RELAXED, __HIP_MEMORY_SCOPE_WORKGROUP);
    __hip_atomic_fetch_add(&lsq[ncol], acc_q, __ATOMIC_RELAXED, __HIP_MEMORY_SCOPE_WORKGROUP);
    __syncthreads();
    if (tid < D) {
        __hip_atomic_fetch_add(gsum   + tid, ls[tid],  __ATOMIC_RELAXED, __HIP_MEMORY_SCOPE_AGENT);
        __hip_atomic_fetch_add(gsumsq + tid, lsq[tid], __ATOMIC_RELAXED, __HIP_MEMORY_SCOPE_AGENT);
    }
}

// ---------------------------------------------------------------------------
// Kernel 5: fold BN stats into per-column scale/shift
// ---------------------------------------------------------------------------
__global__ void k_bn_finalize(const float* __restrict__ gsum,
                              const float* __restrict__ gsumsq,
                              const float* __restrict__ gamma,
                              const float* __restrict__ beta,
                              float* __restrict__ scale,
                              float* __restrict__ shift) {
    int n = threadIdx.x;  // 128 threads, 1 block
    const float inv_n = 1.0f / (float)N_NODES;
    float mean = gsum[n] * inv_n;
    float var  = gsumsq[n] * inv_n - mean * mean;
    float s    = gamma[n] * rsqrtf(var + BN_EPS);
    scale[n] = s;
    shift[n] = beta[n] - mean * s;
}

// ---------------------------------------------------------------------------
// Kernel 6: out = h + scale * rst + shift   (float4 streaming)
// ---------------------------------------------------------------------------
__global__ void k_residual(const float* __restrict__ h,
                           const float* __restrict__ rst,
                           const float* __restrict__ scale,
                           const float* __restrict__ shift,
                           float* __restrict__ out) {
    long vi = (long)blockIdx.x * blockDim.x + threadIdx.x;
    long idx = vi * 4;
    if (idx >= (long)N_NODES * D) return;
    int n = (int)(idx & (D - 1));
    float4 hv = *(const float4*)(h   + idx);
    float4 rv = *(const float4*)(rst + idx);
    float4 sc = *(const float4*)(scale + n);
    float4 sh = *(const float4*)(shift + n);
    float4 o;
    o.x = hv.x + rv.x * sc.x + sh.x;
    o.y = hv.y + rv.y * sc.y + sh.y;
    o.z = hv.z + rv.z * sc.z + sh.z;
    o.w = hv.w + rv.w * sc.w + sh.w;
    *(float4*)(out + idx) = o;
}

// ---------------------------------------------------------------------------
extern "C" void kernel_launch(void* const* d_in, const int* in_sizes, int n_in,
                              void* d_out, int out_size, void* d_ws, size_t ws_size,
                              hipStream_t stream) {
    const float* h      = (const float*)d_in[0];
    const int*   eidx   = (const int*)  d_in[1];
    const float* Wself  = (const float*)d_in[2];
    const float* Wneigh = (const float*)d_in[3];
    const float* bias   = (const float*)d_in[4];
    const float* gamma  = (const float*)d_in[5];
    const float* beta   = (const float*)d_in[6];
    float* out = (float*)d_out;

    // workspace layout (floats)
    float* ws     = (float*)d_ws;
    float* agg    = ws;                            // N*D
    float* deg    = agg    + (size_t)N_NODES * D;  // N
    float* invdeg = deg    + N_NODES;              // N
    float* gsum   = invdeg + N_NODES;              // D
    float* gsumsq = gsum   + D;                    // D
    float* scale  = gsumsq + D;                    // D
    float* shift  = scale  + D;                    // D
    float* rst    = shift  + D;                    // N*D

    // 1) zero agg..sumsq (contiguous: N*D + 2N + 2D floats, divisible by 4)
    long zcount = (long)N_NODES * D + 2L * N_NODES + 2L * D;
    long zvec   = zcount / 4;
    k_zero<<<(unsigned)((zvec + 255) / 256), 256, 0, stream>>>((float4*)ws, zvec);

    // 2) edge scatter: one wave per edge -> 32 threads/edge
    long sthreads = (long)N_EDGES * 32;
    k_scatter<<<(unsigned)((sthreads + 255) / 256), 256, 0, stream>>>(h, eidx, agg, deg);

    // 3) inv_deg
    k_invdeg<<<(N_NODES + 255) / 256, 256, 0, stream>>>(deg, invdeg);

    // 4) fused dual-WMMA GEMM + activation + BN partials
    k_gemm_bn<<<256, 256, 0, stream>>>(h, agg, invdeg, Wself, Wneigh, bias,
                                       rst, gsum, gsumsq);

    // 5) BN finalize
    k_bn_finalize<<<1, D, 0, stream>>>(gsum, gsumsq, gamma, beta, scale, shift);

    // 6) residual output
    long ovec = (long)N_NODES * D / 4;
    k_residual<<<(unsigned)((ovec + 255) / 256), 256, 0, stream>>>(h, rst, scale, shift, out);
}